// NLIF_7318624272612
// MI455X (gfx1250) — compile-verified
//
#include <hip/hip_runtime.h>
#include <hip/hip_bf16.h>

#define NN 8192

typedef __attribute__((ext_vector_type(2))) float v2f;
typedef __attribute__((ext_vector_type(4))) float v4f;
typedef __attribute__((ext_vector_type(8))) float v8f;

// ---------------------------------------------------------------------------
// Kernel 1: y = W @ x for a 16-row block via V_WMMA_F32_16X16X4_F32, with
// x in the A operand (row M=0) and W^T in the B operand so that the 16 row
// results accumulate into D row 0 == c[0] lanes 0..15.
//
// grid = (512, 2): x = row block (16 rows), y = matrix select
// (0: W_syn*s, 1: W_fast*s_fast). block = 256 = 8 waves; each wave owns a
// 1024-wide K slice -> 128 iterations of (1 b128 W-load + 1 b128 x-load +
// 2 WMMAs) covering 8 k-values.
//
// Layout mapping (ISA 7.12.2, 32-bit operands):
//  A 16x4: lane L holds A[m=L&15][K=2*(L>>4)+{0,1}] in {a.x,a.y}.
//    Only m==0 (lanes 0,16) nonzero, carrying x[k'], x[k'+1].
//  B 4x16: lane L holds B[K=2*(L>>4)+{0,1}][n=L&15] in {b.x,b.y}
//    = W[rowbase+n][k'], W[rowbase+n][k'+1].
//  K-pairing: WMMA#1 K0..3 -> {k,k+1,k+4,k+5}; WMMA#2 -> {k+2,k+3,k+6,k+7},
//  so one b128 at (k + 4*(L>>4)) serves both WMMAs (.xy then .zw).
//  D 16x16: row M=0 -> VGPR c[0], lanes 0..15 (N = lane) = partial y.
// ---------------------------------------------------------------------------
__global__ __launch_bounds__(256) void nlif_matvec_wmma(
    const float* __restrict__ Wsyn, const float* __restrict__ Wfast,
    const float* __restrict__ s, const float* __restrict__ sfast,
    float* __restrict__ ws_partials) {
  const int tid  = threadIdx.x;
  const int lane = tid & 31;
  const int wave = tid >> 5;
  const int n    = lane & 15;                 // which row of the 16-row block
  const int hb   = (lane >> 4) << 2;          // 0 or 4: K-halfgroup byte group
  const int rowbase = blockIdx.x << 4;

  const float* __restrict__ W = blockIdx.y ? Wfast : Wsyn;
  const float* __restrict__ x = blockIdx.y ? sfast : s;

  const int k0 = wave << 10;                  // 1024-wide K slice per wave
  const float* Wp = W + (size_t)(rowbase + n) * (size_t)NN + (size_t)(k0 + hb);
  const float* Xp = x + (k0 + hb);
  const bool is_m0 = (n == 0);                // lanes 0 and 16 carry A row 0

  v8f c = {};
  #pragma unroll 8
  for (int it = 0; it < 128; ++it) {          // 8 k-values per iteration
    v4f wv = *(const v4f*)Wp;                 // global_load_b128 (streams W)
    v4f xv = *(const v4f*)Xp;                 // global_load_b128 (cache hit)

    v2f a1, b1, a2, b2;
    a1.x = is_m0 ? xv.x : 0.0f;
    a1.y = is_m0 ? xv.y : 0.0f;
    b1.x = wv.x;
    b1.y = wv.y;
    c = __builtin_amdgcn_wmma_f32_16x16x4_f32(
        false, a1, false, b1, (short)0, c, false, false);

    a2.x = is_m0 ? xv.z : 0.0f;
    a2.y = is_m0 ? xv.w : 0.0f;
    b2.x = wv.z;
    b2.y = wv.w;
    c = __builtin_amdgcn_wmma_f32_16x16x4_f32(
        false, a2, false, b2, (short)0, c, false, false);

    Wp += 8;
    Xp += 8;
  }

  // Fixed-order split-K reduction across the 8 waves (deterministic).
  __shared__ float red[8 * 16];
  if (lane < 16) red[wave * 16 + lane] = c[0];   // D row 0: y[rowbase+lane]
  __syncthreads();
  if (tid < 16) {
    float acc = 0.0f;
    #pragma unroll
    for (int w = 0; w < 8; ++w) acc += red[w * 16 + tid];
    float* o = ws_partials + (blockIdx.y ? NN : 0);
    o[rowbase + tid] = acc;
  }
}

// ---------------------------------------------------------------------------
// Kernel 2: elementwise neuron update. Subtracts the diagonal terms (the
// reference masks the diagonal), adds input current, integrates, resets.
// ---------------------------------------------------------------------------
__global__ __launch_bounds__(256) void nlif_update(
    const float* __restrict__ x_in, const float* __restrict__ v,
    const float* __restrict__ s, const float* __restrict__ sfast,
    const float* __restrict__ W_in, const float* __restrict__ Wsyn,
    const float* __restrict__ Wfast, const float* __restrict__ I_o,
    const float* __restrict__ ws_partials, float* __restrict__ out) {
  const int i = blockIdx.x * 256 + threadIdx.x;
  if (i >= NN) return;

  const float si  = s[i];
  const float sfi = sfast[i];
  const float Isyn  = ws_partials[i]      - Wsyn [(size_t)i * (size_t)(NN + 1)] * si;
  const float Ifast = ws_partials[NN + i] - Wfast[(size_t)i * (size_t)(NN + 1)] * sfi;
  const float Iin   = W_in[2 * i] * x_in[0] + W_in[2 * i + 1] * x_in[1];
  const float Itot  = Isyn + Ifast + Iin + I_o[i];

  const float dv      = Itot / 10.0f;                       // TAU_M
  const float v_next  = v[i] + dv;
  const float gating  = fminf(fmaxf(v_next, 0.0f), 1.0f);
  const float drive   = gating * fminf(fmaxf(dv, -1.0f), 1.0f);
  const float s_new   = si + (drive - si) / 10.0f;          // TAU_S
  const float sf_new  = drive;                              // TAU_S_FAST = 1
  const float spiked  = (v_next >= 1.0f || v_next <= -1.0f) ? 1.0f : 0.0f;
  const float v_new   = (1.0f - spiked) * v_next;

  out[i]                = spiked;               // spiked     [0, 8192)
  out[8194 + i]         = v_new;                // v_new      [8194, +N)
  out[8194 + NN + i]    = s_new;                // s_new
  out[8194 + 2*NN + i]  = sf_new;               // s_fast_new
}

// ---------------------------------------------------------------------------
// Kernel 3: readout partials. 32 blocks x 256 threads; fixed-tree reduction
// of O[0,:]*s_new and O[1,:]*s_new into per-block partials (deterministic).
// ---------------------------------------------------------------------------
__global__ __launch_bounds__(256) void nlif_readout_part(
    const float* __restrict__ O, const float* __restrict__ s_new,
    float* __restrict__ part) {
  const int t = threadIdx.x;
  const int i = blockIdx.x * 256 + t;
  __shared__ float r0[256];
  __shared__ float r1[256];
  const float sv = s_new[i];
  r0[t] = O[i] * sv;
  r1[t] = O[NN + i] * sv;
  __syncthreads();
  for (int off = 128; off > 0; off >>= 1) {
    if (t < off) { r0[t] += r0[t + off]; r1[t] += r1[t + off]; }
    __syncthreads();
  }
  if (t == 0) {
    part[blockIdx.x]      = r0[0];
    part[32 + blockIdx.x] = r1[0];
  }
}

// Kernel 4: final fixed-order sum of the 32 partials per readout row.
__global__ void nlif_readout_final(const float* __restrict__ part,
                                   float* __restrict__ readout) {
  const int r = threadIdx.x;            // 0 or 1
  if (r < 2) {
    float acc = 0.0f;
    for (int b = 0; b < 32; ++b) acc += part[r * 32 + b];
    readout[r] = acc;
  }
}

extern "C" void kernel_launch(void* const* d_in, const int* in_sizes, int n_in,
                              void* d_out, int out_size, void* d_ws, size_t ws_size,
                              hipStream_t stream) {
  const float* x_in  = (const float*)d_in[0];
  const float* v     = (const float*)d_in[1];
  const float* s     = (const float*)d_in[2];
  const float* sfast = (const float*)d_in[3];
  const float* W_in  = (const float*)d_in[4];
  const float* Wsyn  = (const float*)d_in[5];
  const float* Wfast = (const float*)d_in[6];
  const float* O     = (const float*)d_in[7];
  const float* I_o   = (const float*)d_in[8];
  float* out = (float*)d_out;
  float* ws  = (float*)d_ws;   // [0,N): I_syn partial, [N,2N): I_fast partial,
                               // [2N, 2N+64): readout partials

  dim3 g1(NN / 16, 2);
  nlif_matvec_wmma<<<g1, 256, 0, stream>>>(Wsyn, Wfast, s, sfast, ws);
  nlif_update<<<NN / 256, 256, 0, stream>>>(x_in, v, s, sfast, W_in, Wsyn,
                                            Wfast, I_o, ws, out);
  nlif_readout_part<<<NN / 256, 256, 0, stream>>>(O, out + 8194 + NN,
                                                  ws + 2 * NN);
  nlif_readout_final<<<1, 32, 0, stream>>>(ws + 2 * NN, out + NN);
}